// WeightedAdaptiveWingLoss_12206297055683
// MI455X (gfx1250) — compile-verified
//
#include <hip/hip_runtime.h>
#include <hip/hip_bf16.h>
#include <math.h>

// ---------------- problem constants (from reference) ----------------
#define S        160            // spatial size (L = W = H)
#define CCH      8              // channels
#define TS       8              // tile size in L and W (160 = 20*8, exact)
#define HALO     10             // TS + 2 halo rows
#define NROWS    (HALO*HALO)    // 100 rows per tile
#define ROWF4    (S/4)          // 40 float4 per row
#define NF4      (NROWS*ROWF4)  // 4000 float4 loads per tile
#define HALO_EL  (NROWS*S)      // 16000 floats per LDS buffer
#define OUT_EL   (TS*TS*S)      // 10240 outputs per block
#define NTILE    (S/TS)         // 20
#define NBLK     (CCH*NTILE*NTILE) // 3200 blocks
#define INV_N    (1.0f/32768000.0f)

#define GLOBAL_AS __attribute__((address_space(1)))
#define LDS_AS    __attribute__((address_space(3)))

typedef int v4i __attribute__((ext_vector_type(4)));

#if defined(__has_builtin)
# if __has_builtin(__builtin_amdgcn_global_load_async_to_lds_b128)
#  define USE_ASYNC_LDS 1
# endif
# if __has_builtin(__builtin_amdgcn_s_wait_asynccnt)
#  define USE_WAIT_BUILTIN 1
# endif
#endif

__device__ __forceinline__ int refl(int i) {
    i = (i < 0) ? -i : i;                 // -1 -> 1  (jnp.pad 'reflect')
    return (i > S-1) ? (2*(S-1) - i) : i; // 160 -> 158
}

// ------------------------------------------------------------------
// Main kernel: one block = 8x8 (L,W) pencil x full H, one channel.
// Stages y halo tile (10x10x160) in LDS via async global->LDS, does
// separable 3^3 dilation, fused loss, deterministic block reduction.
// ------------------------------------------------------------------
__global__ __launch_bounds__(256) void awing_main(const float* __restrict__ y_pred,
                                                  const float* __restrict__ y,
                                                  float* __restrict__ partial) {
    __shared__ __align__(16) float sY[HALO_EL];  // raw y tile (with halo)
    __shared__ __align__(16) float sH[HALO_EL];  // h-direction 3-max
    __shared__ float swave[8];

    const int t  = threadIdx.x;
    const int c  = blockIdx.z;
    const int l0 = blockIdx.y * TS;
    const int w0 = blockIdx.x * TS;

    // ---- prefetch y_pred tile into cache hierarchy (global_prefetch_b8)
    if (t < 128) {
        int row = t >> 1, half = t & 1;           // 64 rows, 2 segments each
        int l = row / TS, w = row % TS;
        const float* pp = y_pred + (((long)c*S + (l0+l))*S + (w0+w))*S + half*80;
        __builtin_prefetch(pp, 0, 0);
    }

    // ---- stage y halo tile: 100 rows x 160 floats, reflect on L/W edges
    for (int idx = t; idx < NF4; idx += 256) {
        int rr = idx / ROWF4;                     // halo row 0..99
        int f4 = idx - rr*ROWF4;                  // float4 within row
        int li = rr / HALO, wi = rr - li*HALO;
        int gl = refl(l0 + li - 1);
        int gw = refl(w0 + wi - 1);
        long goff = (((long)c*S + gl)*S + gw)*S + f4*4;
        int  loff = rr*S + f4*4;
#if defined(USE_ASYNC_LDS)
        __builtin_amdgcn_global_load_async_to_lds_b128(
            (GLOBAL_AS v4i*)(y + goff),
            (LDS_AS v4i*)(&sY[loff]), 0, 0);
#else
        *(float4*)&sY[loff] = *(const float4*)(y + goff);
#endif
    }
#if defined(USE_ASYNC_LDS)
# if defined(USE_WAIT_BUILTIN)
    __builtin_amdgcn_s_wait_asynccnt(0);
# else
    asm volatile("s_wait_asynccnt 0x0" ::: "memory");
# endif
#endif
    __syncthreads();

    // ---- pass 1: 3-max along H with reflect boundary, for all 100 rows
    for (int e = t; e < HALO_EL; e += 256) {
        int r = e / S, h = e - r*S;
        int base = r * S;
        float v  = sY[e];
        float vm = sY[base + (h == 0    ? 1   : h-1)];
        float vp = sY[base + (h == S-1 ? S-2 : h+1)];
        sH[e] = fmaxf(v, fmaxf(vm, vp));
    }
    __syncthreads();

    // ---- fused (l,w) 3x3 max + loss ; 10240/256 = 40 iters, no divergence
    float acc = 0.0f;
    for (int o = t; o < OUT_EL; o += 256) {
        int l   = o / (TS*S);
        int rem = o - l*(TS*S);
        int w   = rem / S;
        int h   = rem - w*S;
        int rb  = ((l+1)*HALO + (w+1))*S + h;     // interior row in halo frame

        float m0 = fmaxf(sH[rb - HALO*S - S], fmaxf(sH[rb - HALO*S], sH[rb - HALO*S + S]));
        float m1 = fmaxf(sH[rb          - S], fmaxf(sH[rb         ], sH[rb          + S]));
        float m2 = fmaxf(sH[rb + HALO*S - S], fmaxf(sH[rb + HALO*S], sH[rb + HALO*S + S]));
        float dil = fmaxf(m0, fmaxf(m1, m2));
        float Mw  = (dil > 0.2f) ? 1.0f : 0.0f;

        float yv = sY[rb];
        float yp = y_pred[(((long)c*S + (l0+l))*S + (w0+w))*S + h];

        float d  = fabsf(yv - yp);
        float p  = 2.1f - yv;                 // exponent
        float rp = exp2f(-p);                 // (theta/eps)^p with r = 0.5
        float Aa = 28.0f * p * rp / (1.0f + rp);   // omega*(1/(1+rp))*p*2rp/eps
        float Cc = 0.5f * Aa - 14.0f * log1pf(rp);
        float dp = exp2f(p * log2f(d));       // d^p (d=0 -> 0)
        float l1 = 14.0f * log1pf(dp);
        float l2 = Aa * d - Cc;
        float loss = (d < 0.5f) ? l1 : l2;
        acc += loss * (10.0f * Mw + 1.0f);
    }

    // ---- deterministic block reduction (wave32 tree, then 8 wave leaders)
    for (int off = 16; off > 0; off >>= 1)
        acc += __shfl_xor(acc, off, 32);
    if ((t & 31) == 0) swave[t >> 5] = acc;
    __syncthreads();
    if (t == 0) {
        float s = 0.0f;
        #pragma unroll
        for (int i = 0; i < 8; ++i) s += swave[i];
        partial[(blockIdx.z*NTILE + blockIdx.y)*NTILE + blockIdx.x] = s;
    }
}

// ------------------------------------------------------------------
// Final reduction: one full wave (EXEC all ones, as WMMA requires).
// Sum of 32 lane-accumulators via V_WMMA_F32_16X16X4_F32 with B = 1s:
// D[m][n] = rowsum(A, m); lanes 0 and 16 hold the two half totals.
// ------------------------------------------------------------------
typedef float v2f __attribute__((ext_vector_type(2)));
typedef float v8f __attribute__((ext_vector_type(8)));

__global__ __launch_bounds__(32) void awing_reduce(const float* __restrict__ partial,
                                                   float* __restrict__ out) {
    const int lane = threadIdx.x;
    float acc = 0.0f;
    #pragma unroll 4
    for (int j = 0; j < NBLK/32; ++j)        // 3200/32 = 100, fixed order
        acc += partial[lane + 32*j];

#if defined(__has_builtin) && __has_builtin(__builtin_amdgcn_wmma_f32_16x16x4_f32)
    v2f a; a.x = acc; a.y = 0.0f;            // A[16x4]: lane -> (M=lane%16, K=lane<16?0:2)
    v2f b; b.x = 1.0f; b.y = 1.0f;           // B[4x16] all ones
    v8f cd = {};
    cd = __builtin_amdgcn_wmma_f32_16x16x4_f32(
        /*neg_a=*/false, a, /*neg_b=*/false, b,
        /*c_mod=*/(short)0, cd, /*reuse_a=*/false, /*reuse_b=*/false);
    float s = cd[0]+cd[1]+cd[2]+cd[3]+cd[4]+cd[5]+cd[6]+cd[7];
    float tot = __shfl(s, 0, 32) + __shfl(s, 16, 32);
#else
    float s = acc;
    for (int off = 16; off > 0; off >>= 1) s += __shfl_xor(s, off, 32);
    float tot = s;
#endif
    if (lane == 0) out[0] = tot * INV_N;     // mean
}

extern "C" void kernel_launch(void* const* d_in, const int* in_sizes, int n_in,
                              void* d_out, int out_size, void* d_ws, size_t ws_size,
                              hipStream_t stream) {
    (void)in_sizes; (void)n_in; (void)out_size; (void)ws_size;
    const float* y_pred = (const float*)d_in[0];
    const float* y      = (const float*)d_in[1];
    float* out = (float*)d_out;
    float* ws  = (float*)d_ws;                // 3200 partials = 12.8 KB

    dim3 grid(NTILE, NTILE, CCH);             // 20 x 20 x 8 = 3200 blocks
    awing_main  <<<grid, 256, 0, stream>>>(y_pred, y, ws);
    awing_reduce<<<1,    32,  0, stream>>>(ws, out);
}